// SAGEConv_53704271069550
// MI455X (gfx1250) — compile-verified
//
#include <hip/hip_runtime.h>
#include <hip/hip_bf16.h>

typedef __attribute__((ext_vector_type(2))) float v2f;
typedef __attribute__((ext_vector_type(8))) float v8f;

#define D_FEAT 128
#define LDS_STRIDE 132  // 128 + 4 dword pad -> conflict-free ds_load_b64 A-fragments

// ---------------------------------------------------------------------------
// Kernel 1: zero the workspace accumulators (neigh[N,128] and ws[N])
// ---------------------------------------------------------------------------
__global__ __launch_bounds__(256) void zero_ws_kernel(float* __restrict__ p, int n) {
  int i = blockIdx.x * 256 + threadIdx.x;
  if (i < n) p[i] = 0.0f;
}

// ---------------------------------------------------------------------------
// Kernel 2: edge scatter. One wave32 per edge; each lane handles 4 floats.
// All atomics land in L2 (25.8MB accumulator footprint << 192MB L2).
// ---------------------------------------------------------------------------
__global__ __launch_bounds__(256) void edge_scatter_kernel(
    const float* __restrict__ feat, const float* __restrict__ ew,
    const int* __restrict__ src, const int* __restrict__ dst,
    float* __restrict__ neigh, float* __restrict__ ws, int E) {
  int gid  = blockIdx.x * 256 + threadIdx.x;
  int edge = gid >> 5;
  int lane = gid & 31;
  if (edge >= E) return;
  int   s = src[edge];
  int   d = dst[edge];
  float w = ew[edge];
  const float4 f = *(const float4*)(feat + (size_t)s * D_FEAT + lane * 4);
  float* np = neigh + (size_t)d * D_FEAT + lane * 4;
  atomicAdd(np + 0, f.x * w);
  atomicAdd(np + 1, f.y * w);
  atomicAdd(np + 2, f.z * w);
  atomicAdd(np + 3, f.w * w);
  if (lane == 0) atomicAdd(ws + d, w);
}

// ---------------------------------------------------------------------------
// Kernel 3: ws[n] = 1 / max(ws[n], 1)   (in place)
// ---------------------------------------------------------------------------
__global__ __launch_bounds__(256) void recip_ws_kernel(float* __restrict__ ws, int n) {
  int i = blockIdx.x * 256 + threadIdx.x;
  if (i < n) ws[i] = 1.0f / fmaxf(ws[i], 1.0f);
}

// ---------------------------------------------------------------------------
// Kernel 4: fused dual GEMM via V_WMMA_F32_16X16X4_F32.
//   out = 0.5 * (feat @ Wself^T + (neigh*inv_ws) @ Wneigh^T + bself + bneigh)
// Block = 256 threads = 8 waves. Block computes rows [row0, row0+16) x all 128
// cols; wave w computes the 16x16 tile at col0 = 16*w. A tiles (feat + scaled
// neigh) staged once in LDS and shared by all 8 waves.
// Requires N % 16 == 0 (holds: 50000 = 3125*16).
// ---------------------------------------------------------------------------
__global__ __launch_bounds__(256) void sage_gemm_wmma_kernel(
    const float* __restrict__ feat, const float* __restrict__ neigh,
    const float* __restrict__ inv_ws,
    const float* __restrict__ Wself, const float* __restrict__ bself,
    const float* __restrict__ Wneigh, const float* __restrict__ bneigh,
    float* __restrict__ out) {
  __shared__ float lds[2 * 16 * LDS_STRIDE];
  float* lA1 = lds;                     // feat tile      [16][LDS_STRIDE]
  float* lA2 = lds + 16 * LDS_STRIDE;   // h_neigh tile   [16][LDS_STRIDE]

  const int t    = threadIdx.x;
  const int row0 = blockIdx.x * 16;

  // ---- stage A tiles: thread t loads 8 contiguous floats of each matrix ----
  {
    const int r = t >> 4;          // 0..15 row within tile
    const int c = (t & 15) * 8;    // 0..120 col
    const float scale = inv_ws[row0 + r];
    const float4* g1 = (const float4*)(feat  + (size_t)(row0 + r) * D_FEAT + c);
    const float4* g2 = (const float4*)(neigh + (size_t)(row0 + r) * D_FEAT + c);
    float4 x0 = g1[0], x1 = g1[1];
    float4 y0 = g2[0], y1 = g2[1];
    y0.x *= scale; y0.y *= scale; y0.z *= scale; y0.w *= scale;
    y1.x *= scale; y1.y *= scale; y1.z *= scale; y1.w *= scale;
    *(float4*)(lA1 + r * LDS_STRIDE + c)     = x0;
    *(float4*)(lA1 + r * LDS_STRIDE + c + 4) = x1;
    *(float4*)(lA2 + r * LDS_STRIDE + c)     = y0;
    *(float4*)(lA2 + r * LDS_STRIDE + c + 4) = y1;
  }
  __syncthreads();

  // ---- per-lane fragment addressing (CDNA5 WMMA f32 16x16x4 layouts) ----
  const int lane  = t & 31;
  const int wave  = t >> 5;
  const int lsub  = lane & 15;          // row (A) / col (B,C)
  const int khalf = (lane >> 4) << 1;   // 0 for lanes 0-15, 2 for lanes 16-31
  const int col0  = wave * 16;

  const float* a1p = lA1 + lsub * LDS_STRIDE + khalf;
  const float* a2p = lA2 + lsub * LDS_STRIDE + khalf;
  const float* b1p = Wself  + (size_t)(col0 + lsub) * D_FEAT + khalf;  // B[k][n]=W[n][k]
  const float* b2p = Wneigh + (size_t)(col0 + lsub) * D_FEAT + khalf;

  v8f acc = {};
#pragma unroll
  for (int k = 0; k < D_FEAT; k += 4) {
    v2f a1 = *(const v2f*)(a1p + k);
    v2f b1 = *(const v2f*)(b1p + k);
    acc = __builtin_amdgcn_wmma_f32_16x16x4_f32(false, a1, false, b1,
                                                (short)0, acc, false, false);
    v2f a2 = *(const v2f*)(a2p + k);
    v2f b2 = *(const v2f*)(b2p + k);
    acc = __builtin_amdgcn_wmma_f32_16x16x4_f32(false, a2, false, b2,
                                                (short)0, acc, false, false);
  }

  // ---- epilogue: bias + /2, C/D layout: VGPR r -> row r + 8*(lane/16), col lsub
  const int   ocol = col0 + lsub;
  const int   orow = row0 + ((lane >> 4) << 3);
  const float bias = bself[ocol] + bneigh[ocol];
#pragma unroll
  for (int r = 0; r < 8; ++r)
    out[(size_t)(orow + r) * D_FEAT + ocol] = 0.5f * (acc[r] + bias);
}

// ---------------------------------------------------------------------------
extern "C" void kernel_launch(void* const* d_in, const int* in_sizes, int n_in,
                              void* d_out, int out_size, void* d_ws, size_t ws_size,
                              hipStream_t stream) {
  const float* feat   = (const float*)d_in[0];
  const float* ew     = (const float*)d_in[1];
  const int*   src    = (const int*)d_in[2];
  const int*   dst    = (const int*)d_in[3];
  const float* Wself  = (const float*)d_in[4];
  const float* bself  = (const float*)d_in[5];
  const float* Wneigh = (const float*)d_in[6];
  const float* bneigh = (const float*)d_in[7];
  float*       out    = (float*)d_out;

  const int N = in_sizes[0] / D_FEAT;  // 50000 (multiple of 16)
  const int E = in_sizes[1];           // 800000

  // workspace layout: neigh[N*128] | ws[N]
  float* neigh = (float*)d_ws;
  float* ws    = neigh + (size_t)N * D_FEAT;

  const int nz = N * D_FEAT + N;
  zero_ws_kernel<<<(nz + 255) / 256, 256, 0, stream>>>(neigh, nz);

  // one wave32 per edge -> E*32 threads
  const long long sthreads = (long long)E * 32;
  edge_scatter_kernel<<<(int)((sthreads + 255) / 256), 256, 0, stream>>>(
      feat, ew, src, dst, neigh, ws, E);

  recip_ws_kernel<<<(N + 255) / 256, 256, 0, stream>>>(ws, N);

  sage_gemm_wmma_kernel<<<N / 16, 256, 0, stream>>>(
      feat, neigh, ws, Wself, bself, Wneigh, bneigh, out);
}